// SPConvlayer_38757784879441
// MI455X (gfx1250) — compile-verified
//
#include <hip/hip_runtime.h>

typedef __attribute__((ext_vector_type(2))) float v2f;
typedef __attribute__((ext_vector_type(8))) float v8f;

#define NB   4096
#define BB   8
#define KDIM 128   // M*A
#define JDIM 64    // P*W

// h workspace layout (pair-interleaved over the row index n):
//   h[b, n, col]  ->  hws[b*NB*64 + (n>>1)*128 + col*2 + (n&1)]
// so the two K-values a WMMA B-fragment lane needs are adjacent (one b64 load).

// ---------------------------------------------------------------------------
// Kernel 1: h[b,n,j] = sum_k x[b,n,k] * W2[k,j],  k=m*32+a, j=p*16+w
// 32 rows per block, 64 threads (2 waves), f32 WMMA 16x16x4.
// ---------------------------------------------------------------------------
__global__ __launch_bounds__(64) void spconv_feat_kernel(
    const float* __restrict__ x, const float* __restrict__ weight,
    float* __restrict__ h) {
  __shared__ float sW[64 * 160];    // pair-interleaved: [k2][j*2 + parity], 40.0 KB
  __shared__ float sX[32 * 132];    // stride 132 -> bank spread for A frags

  const int tid  = threadIdx.x;
  const int wave = tid >> 5;
  const int lane = tid & 31;
  const int l16  = lane & 15;
  const int hi   = lane >> 4;

  // stage reordered + pair-interleaved weight
  for (int i = tid; i < KDIM * JDIM; i += 64) {
    const int k = i >> 6, j = i & 63;
    const int m = k >> 5, a = k & 31, p = j >> 4, w = j & 15;
    sW[(k >> 1) * 160 + j * 2 + (k & 1)] = weight[(((m * 4 + p) * 32) + a) * 16 + w];
  }
  // stage 32 contiguous x rows (one contiguous 16KB region)
  const float4* xblk = (const float4*)(x + (size_t)blockIdx.x * (32 * KDIM));
  for (int i = tid; i < 32 * 32; i += 64) {
    const int row = i >> 5, c4 = i & 31;
    *(float4*)&sX[row * 132 + c4 * 4] = xblk[i];
  }
  __syncthreads();

  v8f acc[4] = {};
  const int rb = wave * 16 + l16;
#pragma unroll
  for (int kk = 0; kk < KDIM; kk += 4) {
    const int kL = kk + 2 * hi;
    const int k2 = (kk >> 1) + hi;
    const v2f a = *(const v2f*)&sX[rb * 132 + kL];
#pragma unroll
    for (int t = 0; t < 4; ++t) {
      const v2f bb = *(const v2f*)&sW[k2 * 160 + (t * 16 + l16) * 2];
      acc[t] = __builtin_amdgcn_wmma_f32_16x16x4_f32(
          false, a, false, bb, (short)0, acc[t], false, false);
    }
  }

  // store h pair-interleaved: block covers rows [blk*32, blk*32+32) = 16 pairs
  float* hblk = h + (size_t)blockIdx.x * (32 * JDIM);
#pragma unroll
  for (int t = 0; t < 4; ++t) {
    const int col = t * 16 + l16;
#pragma unroll
    for (int v = 0; v < 8; ++v) {
      const int rl = wave * 16 + v + 8 * hi;
      hblk[(rl >> 1) * 128 + col * 2 + (rl & 1)] = acc[t][v];
    }
  }
}

// ---------------------------------------------------------------------------
// Kernel 2: per batch b:  out[n,:] = relu( rinv[n]*(adj[n,:]@h + h[n,:]) ) + bias
// 128 rows/WG, 128 threads (4 waves), KC=64 chunks, register-prefetch pipeline.
// Rowsum of adj accumulated for free during global loads.
// ---------------------------------------------------------------------------
#define KC 64

__device__ __forceinline__ void load_chunk(const float* __restrict__ adjB,
                                           const float* __restrict__ hB,
                                           int kc, int tid,
                                           float4* ra, float4* rh, float* rs) {
  const int rbase = tid >> 4, c4 = tid & 15;
#pragma unroll
  for (int i = 0; i < 16; ++i) {
    const int row = rbase + 8 * i;
    const float4 v = *(const float4*)(adjB + (size_t)row * NB + kc + c4 * 4);
    ra[i] = v;
    rs[i] += v.x + v.y + v.z + v.w;
  }
  // h chunk: rows [kc, kc+64) == pairs [kc/2, kc/2+32) == contiguous 4096 floats
  const float4* hChunk = (const float4*)(hB + (size_t)kc * JDIM);
#pragma unroll
  for (int i = 0; i < 8; ++i)
    rh[i] = hChunk[tid + 128 * i];
}

__device__ __forceinline__ void store_chunk(float* sAdj, float* sH, int tid,
                                            const float4* ra, const float4* rh) {
  const int rbase = tid >> 4, c4 = tid & 15;
#pragma unroll
  for (int i = 0; i < 16; ++i)
    *(float4*)&sAdj[(rbase + 8 * i) * 68 + c4 * 4] = ra[i];
  // interleaved h: each k2-row is 128 floats = 32 float4s; LDS stride 160 floats
#pragma unroll
  for (int i = 0; i < 8; ++i) {
    const int j = tid + 128 * i;
    *(float4*)&sH[(j >> 5) * 160 + (j & 31) * 4] = rh[i];
  }
}

__global__ __launch_bounds__(128) void spconv_agg_kernel(
    const float* __restrict__ adj, const float* __restrict__ h,
    const float* __restrict__ bias, float* __restrict__ out) {
  __shared__ float sAdj[128 * 68];     // 34.0 KB, stride 68: A-frag b64 conflict-free
  __shared__ float sH[(KC / 2) * 160]; // 20.0 KB, pair-interleaved, b64 conflict-free
  __shared__ float sRinv[128];

  const int tid  = threadIdx.x;
  const int wave = tid >> 5;
  const int lane = tid & 31;
  const int l16  = lane & 15;
  const int hi   = lane >> 4;
  const int b      = blockIdx.y;
  const int rowblk = blockIdx.x * 128;

  const float* adjB = adj + ((size_t)b * NB + rowblk) * NB;
  const float* hB   = h + (size_t)b * NB * JDIM;

  float rs[16];
#pragma unroll
  for (int i = 0; i < 16; ++i) rs[i] = 0.0f;

  float4 ra[16], rh[8];
  load_chunk(adjB, hB, 0, tid, ra, rh, rs);
  store_chunk(sAdj, sH, tid, ra, rh);
  __syncthreads();

  v8f acc[2][4] = {};
  const int r0 = (wave * 32 + l16) * 68;
  const int r1 = (wave * 32 + 16 + l16) * 68;

  for (int kc = 0; kc < NB; kc += KC) {
    const bool more = (kc + KC) < NB;
    if (more) load_chunk(adjB, hB, kc + KC, tid, ra, rh, rs);  // prefetch next
#pragma unroll
    for (int kk = 0; kk < KC; kk += 4) {
      const int kL = kk + 2 * hi;
      const int k2 = (kk >> 1) + hi;
      const v2f a0 = *(const v2f*)&sAdj[r0 + kL];
      const v2f a1 = *(const v2f*)&sAdj[r1 + kL];
#pragma unroll
      for (int t = 0; t < 4; ++t) {
        const v2f bb = *(const v2f*)&sH[k2 * 160 + (t * 16 + l16) * 2];
        acc[0][t] = __builtin_amdgcn_wmma_f32_16x16x4_f32(
            false, a0, false, bb, (short)0, acc[0][t], false, false);
        acc[1][t] = __builtin_amdgcn_wmma_f32_16x16x4_f32(
            false, a1, false, bb, (short)0, acc[1][t], false, false);
      }
    }
    __syncthreads();
    if (more) store_chunk(sAdj, sH, tid, ra, rh);
    __syncthreads();
  }

  // rowsum reduction: thread t holds partials for rows (t>>4) + 8i
#pragma unroll
  for (int i = 0; i < 16; ++i) sAdj[tid * 16 + i] = rs[i];
  __syncthreads();
  {
    float s = 1.0f;  // +I self-loop contributes 1 to the rowsum
#pragma unroll
    for (int c = 0; c < 16; ++c)
      s += sAdj[(((tid & 7) * 16) + c) * 16 + (tid >> 3)];
    sRinv[tid] = 1.0f / s;
  }
  __syncthreads();

  float* outB = out + ((size_t)b * NB + rowblk) * JDIM;
#pragma unroll
  for (int s = 0; s < 2; ++s) {
    const int rbase = wave * 32 + s * 16;
#pragma unroll
    for (int t = 0; t < 4; ++t) {
      const int col = t * 16 + l16;
      const float bv = bias[col];
#pragma unroll
      for (int v = 0; v < 8; ++v) {
        const int rl = rbase + v + 8 * hi;
        const int n = rowblk + rl;
        float val = acc[s][t][v] + hB[(size_t)(n >> 1) * 128 + col * 2 + (n & 1)];
        val *= sRinv[rl];
        val = fmaxf(val, 0.0f) + bv;
        outB[(size_t)rl * JDIM + col] = val;
      }
    }
  }
}

// ---------------------------------------------------------------------------
extern "C" void kernel_launch(void* const* d_in, const int* in_sizes, int n_in,
                              void* d_out, int out_size, void* d_ws, size_t ws_size,
                              hipStream_t stream) {
  const float* x      = (const float*)d_in[0];  // [8,4096,4,32]
  const float* adj    = (const float*)d_in[1];  // [8,4096,4096]
  const float* weight = (const float*)d_in[2];  // [4,4,32,16]
  const float* bias   = (const float*)d_in[3];  // [1,1,4,16]
  float* out = (float*)d_out;                   // [8,4096,4,16]
  float* hbuf = (float*)d_ws;                   // needs 8*4096*64*4 = 8.4 MB

  (void)in_sizes; (void)n_in; (void)out_size; (void)ws_size;

  spconv_feat_kernel<<<dim3((BB * NB) / 32), 64, 0, stream>>>(x, weight, hbuf);
  spconv_agg_kernel<<<dim3(NB / 128, BB), 128, 0, stream>>>(adj, hbuf, bias, out);
}